// RelGraphEmbedLayer_34368328303102
// MI455X (gfx1250) — compile-verified
//
#include <hip/hip_runtime.h>

typedef __attribute__((ext_vector_type(2))) float v2f;
typedef __attribute__((ext_vector_type(4))) float v4f;
typedef __attribute__((ext_vector_type(8))) float v8f;

constexpr int kN = 131072;  // minibatch nodes
constexpr int kD = 512;     // raw feature dim
constexpr int kE = 256;     // embedding size

constexpr int BLOCK_M  = 64;       // nodes per workgroup (output tile rows)
constexpr int KC       = 16;       // K chunk staged in LDS per iteration
constexpr int A_STRIDE = KC + 4;   // 20 floats: 16B-aligned rows, conflict-free b64 frag loads
constexpr int KP       = KC / 2;   // K pair-rows per chunk
constexpr int W_PITCH  = kE + 16;  // 272 float2 per pair-row: mod 32 == 16 -> conflict-free b64

// y[n] = (tid==0) feats0[tix]@W0  +  (tid==1) feats1[tix]@W1  +  (tid>=2) node_emb[nid]
__global__ __launch_bounds__(256)
void relgraph_embed_wmma(const int* __restrict__ node_ids,
                         const int* __restrict__ node_tids,
                         const int* __restrict__ type_ids,
                         const float* __restrict__ feats0,
                         const float* __restrict__ feats1,
                         const float* __restrict__ W0,
                         const float* __restrict__ W1,
                         const float* __restrict__ node_emb,
                         float* __restrict__ out)
{
    __shared__ float lsA0[BLOCK_M * A_STRIDE];   // type-0 masked gathered feats chunk
    __shared__ float lsA1[BLOCK_M * A_STRIDE];   // type-1 masked gathered feats chunk
    // W chunks, fragment-major: pair-row pr, col c -> float2(W[k0+2pr][c], W[k0+2pr+1][c])
    __shared__ float lsW0[KP * W_PITCH * 2];
    __shared__ float lsW1[KP * W_PITCH * 2];
    __shared__ int   ls_tid[BLOCK_M];
    __shared__ int   ls_nid[BLOCK_M];
    __shared__ int   ls_tix[BLOCK_M];

    const int tid  = threadIdx.x;
    const int wave = tid >> 5;
    const int lane = tid & 31;
    const int lm   = lane & 15;   // lane within half-wave
    const int hi   = lane >> 4;   // half-wave select (K split for A/B frags)
    const int mb   = wave & 3;    // M band: 16 rows
    const int nh   = wave >> 2;   // N half: 128 cols (8 tiles of 16)
    const long rbase = (long)blockIdx.x * BLOCK_M;

    if (tid < BLOCK_M) {
        int t = node_tids[rbase + tid];
        ls_tid[tid] = t;
        ls_nid[tid] = node_ids[rbase + tid];
        ls_tix[tid] = (t <= 1) ? type_ids[rbase + tid] : 0;
    }

    v8f acc[8] = {};

    // A gather: 64 rows x KC floats = 1024 floats / 256 thr -> one v4f each
    const int gr = tid >> 2;
    const int gc = (tid & 3) * 4;
    // W stage: 8 pair-rows x 256 cols of float2 / 256 thr -> 8 float2 each
    const int pr = tid >> 5;          // pair-row 0..7
    const int cs = (tid & 31) * 8;    // 8 columns

    for (int k0 = 0; k0 < kD; k0 += KC) {
        __syncthreads();  // protect LDS reuse (also covers metadata on iter 0)

        // ---- gather A chunk (type-masked), NT: feats rows are read exactly once ----
        {
            int t = ls_tid[gr];
            const float* srcTab = (t == 0) ? feats0 : feats1;
            const v4f* src = (const v4f*)(srcTab + (size_t)ls_tix[gr] * kD + k0 + gc);
            v4f z = {0.0f, 0.0f, 0.0f, 0.0f};
            v4f v = z;
            if (t <= 1) v = __builtin_nontemporal_load(src);
            *(v4f*)&lsA0[gr * A_STRIDE + gc] = (t == 0) ? v : z;
            *(v4f*)&lsA1[gr * A_STRIDE + gc] = (t == 1) ? v : z;
        }
        // ---- stage W chunk pair-interleaved (temporal: W0/W1 stay L2-resident) ----
        {
            const float* wa = W0 + (size_t)(k0 + 2 * pr) * kE + cs;   // row k
            const float* wb = wa + kE;                                 // row k+1
            v4f a0 = *(const v4f*)(wa);
            v4f a1 = *(const v4f*)(wa + 4);
            v4f b0 = *(const v4f*)(wb);
            v4f b1 = *(const v4f*)(wb + 4);
            float* dst = &lsW0[(pr * W_PITCH + cs) * 2];
            v4f i0 = {a0.x, b0.x, a0.y, b0.y};
            v4f i1 = {a0.z, b0.z, a0.w, b0.w};
            v4f i2 = {a1.x, b1.x, a1.y, b1.y};
            v4f i3 = {a1.z, b1.z, a1.w, b1.w};
            *(v4f*)(dst)      = i0;
            *(v4f*)(dst + 4)  = i1;
            *(v4f*)(dst + 8)  = i2;
            *(v4f*)(dst + 12) = i3;

            const float* va = W1 + (size_t)(k0 + 2 * pr) * kE + cs;
            const float* vb = va + kE;
            a0 = *(const v4f*)(va);
            a1 = *(const v4f*)(va + 4);
            b0 = *(const v4f*)(vb);
            b1 = *(const v4f*)(vb + 4);
            dst = &lsW1[(pr * W_PITCH + cs) * 2];
            v4f j0 = {a0.x, b0.x, a0.y, b0.y};
            v4f j1 = {a0.z, b0.z, a0.w, b0.w};
            v4f j2 = {a1.x, b1.x, a1.y, b1.y};
            v4f j3 = {a1.z, b1.z, a1.w, b1.w};
            *(v4f*)(dst)      = j0;
            *(v4f*)(dst + 4)  = j1;
            *(v4f*)(dst + 8)  = j2;
            *(v4f*)(dst + 12) = j3;
        }
        __syncthreads();

        const float* ar0 = &lsA0[(mb * 16 + lm) * A_STRIDE];
        const float* ar1 = &lsA1[(mb * 16 + lm) * A_STRIDE];
        const int cbase = nh * 128 + lm;
#pragma unroll
        for (int kk = 0; kk < KC; kk += 4) {
            // A frag 16x4 f32: lanes 0-15 hold M, VGPR pair holds K={kk+2hi, kk+2hi+1}
            v2f a0 = *(const v2f*)(ar0 + kk + 2 * hi);
            v2f a1 = *(const v2f*)(ar1 + kk + 2 * hi);
            // B frag: pair-row (kk/2 + hi) holds K={kk+2hi, kk+2hi+1} as one float2
            const int prow = (kk >> 1) + hi;
            const float* b0base = &lsW0[(prow * W_PITCH + cbase) * 2];
            const float* b1base = &lsW1[(prow * W_PITCH + cbase) * 2];
#pragma unroll
            for (int t = 0; t < 8; ++t) {
                v2f b0 = *(const v2f*)(b0base + t * 32);
                v2f b1 = *(const v2f*)(b1base + t * 32);
                acc[t] = __builtin_amdgcn_wmma_f32_16x16x4_f32(
                    false, a0, false, b0, (short)0, acc[t], false, false);
                acc[t] = __builtin_amdgcn_wmma_f32_16x16x4_f32(
                    false, a1, false, b1, (short)0, acc[t], false, false);
            }
        }
    }

    // ---- epilogue: C layout is VGPR i, half hi -> M = i + 8*hi, N = lm ----
#pragma unroll
    for (int t = 0; t < 8; ++t) {
        const int col = nh * 128 + t * 16 + lm;
#pragma unroll
        for (int i = 0; i < 8; ++i) {
            const int rloc = mb * 16 + i + 8 * hi;
            float v = acc[t][i];
            if (ls_tid[rloc] >= 2) {
                v += __builtin_nontemporal_load(
                    &node_emb[(size_t)ls_nid[rloc] * kE + col]);
            }
            __builtin_nontemporal_store(v, &out[(rbase + rloc) * (long)kE + col]);
        }
    }
}

extern "C" void kernel_launch(void* const* d_in, const int* in_sizes, int n_in,
                              void* d_out, int out_size, void* d_ws, size_t ws_size,
                              hipStream_t stream)
{
    const int*   node_ids  = (const int*)d_in[0];
    const int*   node_tids = (const int*)d_in[1];
    const int*   type_ids  = (const int*)d_in[2];
    const float* feats0    = (const float*)d_in[3];
    const float* feats1    = (const float*)d_in[4];
    const float* W0        = (const float*)d_in[5];
    const float* W1        = (const float*)d_in[6];
    const float* node_emb  = (const float*)d_in[7];
    float* out = (float*)d_out;

    dim3 grid(kN / BLOCK_M);
    dim3 block(256);
    hipLaunchKernelGGL(relgraph_embed_wmma, grid, block, 0, stream,
                       node_ids, node_tids, type_ids,
                       feats0, feats1, W0, W1, node_emb, out);
}